// ExpertC_27530740367431
// MI455X (gfx1250) — compile-verified
//
#include <hip/hip_runtime.h>
#include <math.h>

// ---------------- model constants ----------------
#define INPUT_DIM 1024
#define HIDDEN    256
#define OUT_DIM   256
#define NLAYERS   2
#define D_STATE   32
#define HEADDIM   32
#define D_INNER   512              // EXPAND*HIDDEN
#define NHEADS    16               // D_INNER/HEADDIM
#define CONV_DIM  576              // D_INNER + 2*D_STATE
#define D_IN_PROJ 1104             // 2*D_INNER + 2*D_STATE + NHEADS
#define KCONV     4
#define BB        16
#define LL        2048
#define TOKENS    (BB*LL)          // 32768
#define QCHUNK    64               // scan chunk length
#define NCHUNK    (LL/QCHUNK)      // 32
#define NCID      (BB*NHEADS*NCHUNK) // 8192
#define EPSF      1e-5f

typedef float v2f __attribute__((ext_vector_type(2)));
typedef float v8f __attribute__((ext_vector_type(8)));

#define WMMA4(a, b, c) \
    __builtin_amdgcn_wmma_f32_16x16x4_f32(false, (a), false, (b), (short)0, (c), false, false)

// ---------------- helpers ----------------
__device__ __forceinline__ float siluf(float x) { return x / (1.0f + expf(-x)); }
__device__ __forceinline__ float softplusf_(float x) {
    return (x > 20.0f) ? x : log1pf(expf(x));
}
__device__ __forceinline__ float geluf(float x) {
    return 0.5f * x * (1.0f + erff(x * 0.70710678118654752f));
}
__device__ __forceinline__ float waveRedSum(float v) {
#pragma unroll
    for (int m = 16; m >= 1; m >>= 1) v += __shfl_xor(v, m, 32);
    return v;
}

// ---------------- WMMA fp32 GEMM: C[M,N] = act(A[M,K] @ W[K,N] + bias[N]) ----------------
// block = 128 threads (4 waves), block tile 64x64, K chunked by 32 via LDS.
// Requires M%64==0 and K%32==0 (true for all GEMMs here); N guarded.
template <int ACT>  // 0 = none, 1 = exact gelu
__global__ __launch_bounds__(128)
void gemm_wmma_f32(const float* __restrict__ A, const float* __restrict__ W,
                   const float* __restrict__ bias, float* __restrict__ C,
                   int M, int N, int K) {
    __shared__ float As[32][65];   // As[k][m]
    __shared__ float Bs[32][65];   // Bs[k][n]

    const int tid  = threadIdx.x;
    const int lane = tid & 31;
    const int wave = tid >> 5;
    const int lh   = lane & 15;
    const int hi   = lane >> 4;

    const int rowBlock = blockIdx.y << 6;
    const int nBlock   = blockIdx.x << 6;

    v8f acc0 = {}, acc1 = {}, acc2 = {}, acc3 = {};

    const int ak     = tid & 31;
    const int amBase = tid >> 5;
    const int bn     = tid & 63;
    const int bkBase = tid >> 6;
    const int bcol   = nBlock + bn;
    const bool bok   = (bcol < N);

    for (int k0 = 0; k0 < K; k0 += 32) {
        if (k0 + 32 < K)  // L2 prefetch of next A chunk (global_prefetch_b8)
            __builtin_prefetch(&A[(size_t)(rowBlock + amBase) * K + (k0 + 32 + ak)], 0, 3);
#pragma unroll 4
        for (int m = amBase; m < 64; m += 4)
            As[ak][m] = A[(size_t)(rowBlock + m) * K + (k0 + ak)];
#pragma unroll 4
        for (int k = bkBase; k < 32; k += 2)
            Bs[k][bn] = bok ? W[(size_t)(k0 + k) * N + bcol] : 0.0f;
        __syncthreads();

#pragma unroll
        for (int kk = 0; kk < 32; kk += 4) {
            v2f a;
            a[0] = As[kk + hi * 2 + 0][(wave << 4) + lh];
            a[1] = As[kk + hi * 2 + 1][(wave << 4) + lh];
            v2f b0, b1, b2, b3;
            b0[0] = Bs[kk + hi * 2 + 0][0  + lh]; b0[1] = Bs[kk + hi * 2 + 1][0  + lh];
            b1[0] = Bs[kk + hi * 2 + 0][16 + lh]; b1[1] = Bs[kk + hi * 2 + 1][16 + lh];
            b2[0] = Bs[kk + hi * 2 + 0][32 + lh]; b2[1] = Bs[kk + hi * 2 + 1][32 + lh];
            b3[0] = Bs[kk + hi * 2 + 0][48 + lh]; b3[1] = Bs[kk + hi * 2 + 1][48 + lh];
            acc0 = WMMA4(a, b0, acc0);
            acc1 = WMMA4(a, b1, acc1);
            acc2 = WMMA4(a, b2, acc2);
            acc3 = WMMA4(a, b3, acc3);
        }
        __syncthreads();
    }

    const int rowBase = rowBlock + (wave << 4) + (hi << 3);
#pragma unroll
    for (int j = 0; j < 4; ++j) {
        const v8f acc = (j == 0) ? acc0 : (j == 1) ? acc1 : (j == 2) ? acc2 : acc3;
        const int col = nBlock + (j << 4) + lh;
        if (col < N) {
            const float bv = bias[col];
#pragma unroll
            for (int r = 0; r < 8; ++r) {
                float v = acc[r] + bv;
                if (ACT == 1) v = geluf(v);
                C[(size_t)(rowBase + r) * N + col] = v;
            }
        }
    }
}

// ---------------- depthwise causal conv (K=4) + SiLU ----------------
__global__ void conv_silu_kernel(const float* __restrict__ zx,
                                 const float* __restrict__ convw,
                                 const float* __restrict__ convb,
                                 float* __restrict__ xbc) {
    const size_t idx = (size_t)blockIdx.x * blockDim.x + threadIdx.x;
    if (idx >= (size_t)TOKENS * CONV_DIM) return;
    const int c   = (int)(idx % CONV_DIM);
    const int tok = (int)(idx / CONV_DIM);
    const int b = tok / LL, t = tok % LL;
    float acc = convb[c];
#pragma unroll
    for (int i = 0; i < KCONV; ++i) {
        const int tt = t - (KCONV - 1) + i;
        if (tt >= 0)
            acc += zx[(size_t)(b * LL + tt) * D_IN_PROJ + D_INNER + c] * convw[c * KCONV + i];
    }
    xbc[(size_t)tok * CONV_DIM + c] = siluf(acc);
}

// ---------------- dt = softplus(raw + dtb) ----------------
__global__ void dt_softplus_kernel(const float* __restrict__ zx,
                                   const float* __restrict__ dtb,
                                   float* __restrict__ dtOut) {
    const int idx = blockIdx.x * blockDim.x + threadIdx.x;
    if (idx >= TOKENS * NHEADS) return;
    const int h = idx & (NHEADS - 1);
    const int tok = idx >> 4;
    const float raw = zx[(size_t)tok * D_IN_PROJ + (D_INNER + CONV_DIM) + h];
    dtOut[idx] = softplusf_(raw + dtb[h]);
}

// =====================================================================
// Chunked SSD scan (Mamba-2 style): serial depth 32 instead of 2048.
// cid = ((b*NHEADS)+h)*NCHUNK + c
// =====================================================================

// ---- Phase A: per-chunk intra terms + chunk-local states (all WMMA) ----
__global__ __launch_bounds__(128)
void chunk_intra_kernel(const float* __restrict__ xbc,   // [TOKENS, CONV_DIM]
                        const float* __restrict__ dtbuf, // [TOKENS, NHEADS]
                        const float* __restrict__ Alog,  // [NHEADS]
                        const float* __restrict__ Dp,    // [NHEADS]
                        float* __restrict__ ybuf,        // [TOKENS, D_INNER]
                        float* __restrict__ hc_buf,      // [NCID, 32, 32]  (HcT[n][p])
                        float* __restrict__ cume_buf,    // [NCID, 64]      exp(cum_t)
                        float* __restrict__ lam_buf) {   // [NCID]          exp(cum_63)
    __shared__ float Xc[QCHUNK][33];  // x[s][p]
    __shared__ float Bc[QCHUNK][33];  // B[s][n]
    __shared__ float Cc[QCHUNK][33];  // C[t][n]
    __shared__ float Gs[QCHUNK][65];  // masked decay * CB
    __shared__ float cum[QCHUNK], dtl[QCHUNK], wl[QCHUNK];

    const int cid = blockIdx.x;
    const int c   = cid & (NCHUNK - 1);
    const int h   = (cid >> 5) & (NHEADS - 1);
    const int b   = cid >> 9;
    const int tokBase = b * LL + c * QCHUNK;

    const int tid  = threadIdx.x;
    const int lane = tid & 31;
    const int wave = tid >> 5;
    const int lh   = lane & 15;
    const int hi   = lane >> 4;

    const float A  = -expf(Alog[h]);
    const float Dv = Dp[h];

    // ---- stage chunk data ----
    for (int s = wave; s < QCHUNK; s += 4) {
        const float* row = xbc + (size_t)(tokBase + s) * CONV_DIM;
        Xc[s][lane] = row[h * HEADDIM + lane];
        Bc[s][lane] = row[D_INNER + lane];
        Cc[s][lane] = row[D_INNER + D_STATE + lane];
    }
    if (tid < QCHUNK) dtl[tid] = dtbuf[(size_t)(tokBase + tid) * NHEADS + h];
    __syncthreads();

    // ---- inclusive prefix sum of a_t = dt_t * A (Hillis-Steele) ----
    if (tid < QCHUNK) cum[tid] = dtl[tid] * A;
    __syncthreads();
    for (int off = 1; off < QCHUNK; off <<= 1) {
        float add = 0.0f;
        if (tid < QCHUNK && tid >= off) add = cum[tid - off];
        __syncthreads();
        if (tid < QCHUNK) cum[tid] += add;
        __syncthreads();
    }
    if (tid < QCHUNK) {
        wl[tid] = expf(cum[QCHUNK - 1] - cum[tid]) * dtl[tid];
        cume_buf[(size_t)cid * QCHUNK + tid] = expf(cum[tid]);
    }
    if (tid == 0) lam_buf[cid] = expf(cum[QCHUNK - 1]);
    __syncthreads();

    // ---- GEMM1: CB[t][s] = sum_n C[t][n]*B[s][n] (64x64, K=32) ----
    v8f accG[4] = {v8f{}, v8f{}, v8f{}, v8f{}};
#pragma unroll
    for (int kk = 0; kk < 32; kk += 4) {
        const int k0 = kk + hi * 2, k1 = k0 + 1;
        v2f a; a[0] = Cc[(wave << 4) + lh][k0]; a[1] = Cc[(wave << 4) + lh][k1];
#pragma unroll
        for (int j = 0; j < 4; ++j) {
            v2f bb; bb[0] = Bc[(j << 4) + lh][k0]; bb[1] = Bc[(j << 4) + lh][k1];
            accG[j] = WMMA4(a, bb, accG[j]);
        }
    }
    // mask + decay -> Gs
#pragma unroll
    for (int j = 0; j < 4; ++j) {
        const int s = (j << 4) + lh;
#pragma unroll
        for (int r = 0; r < 8; ++r) {
            const int t = (wave << 4) + (hi << 3) + r;
            Gs[t][s] = (s <= t) ? accG[j][r] * expf(cum[t] - cum[s]) : 0.0f;
        }
    }
    __syncthreads();

    // ---- GEMM2: Y_intra = G @ (dt*X) (64x32, K=64) ; + D*x ----
    v8f accY[2] = {v8f{}, v8f{}};
#pragma unroll
    for (int kk = 0; kk < QCHUNK; kk += 4) {
        const int s0 = kk + hi * 2, s1 = s0 + 1;
        v2f a; a[0] = Gs[(wave << 4) + lh][s0]; a[1] = Gs[(wave << 4) + lh][s1];
        const float d0 = dtl[s0], d1 = dtl[s1];
#pragma unroll
        for (int j = 0; j < 2; ++j) {
            v2f bb; bb[0] = Xc[s0][(j << 4) + lh] * d0; bb[1] = Xc[s1][(j << 4) + lh] * d1;
            accY[j] = WMMA4(a, bb, accY[j]);
        }
    }
#pragma unroll
    for (int j = 0; j < 2; ++j) {
        const int p = (j << 4) + lh;
#pragma unroll
        for (int r = 0; r < 8; ++r) {
            const int t = (wave << 4) + (hi << 3) + r;
            ybuf[(size_t)(tokBase + t) * D_INNER + h * HEADDIM + p] =
                accY[j][r] + Dv * Xc[t][p];
        }
    }

    // ---- GEMM3: HcT[n][p] = sum_s (B[s][n]*w_s)*X[s][p] (32x32, K=64) ----
    const int rn = (wave >> 1) << 4;   // row (state n) tile
    const int cp = (wave & 1) << 4;    // col (headdim p) tile
    v8f accH = {};
#pragma unroll
    for (int kk = 0; kk < QCHUNK; kk += 4) {
        const int s0 = kk + hi * 2, s1 = s0 + 1;
        v2f a;  a[0]  = Bc[s0][rn + lh] * wl[s0]; a[1]  = Bc[s1][rn + lh] * wl[s1];
        v2f bb; bb[0] = Xc[s0][cp + lh];          bb[1] = Xc[s1][cp + lh];
        accH = WMMA4(a, bb, accH);
    }
#pragma unroll
    for (int r = 0; r < 8; ++r) {
        const int n = rn + (hi << 3) + r;
        hc_buf[(size_t)cid * (D_STATE * HEADDIM) + n * HEADDIM + (cp + lh)] = accH[r];
    }
}

// ---- Phase B: serial chunk-state propagation (256 waves x 32 steps) ----
__global__ __launch_bounds__(32)
void chunk_seq_kernel(const float* __restrict__ hc_buf,
                      const float* __restrict__ lam_buf,
                      float* __restrict__ hprev_buf) {
    const int bh = blockIdx.x;       // b*NHEADS + h
    const int p  = threadIdx.x;      // headdim lane
    float hst[D_STATE];
#pragma unroll
    for (int n = 0; n < D_STATE; ++n) hst[n] = 0.0f;
    for (int c = 0; c < NCHUNK; ++c) {
        const size_t base = ((size_t)bh * NCHUNK + c) * (D_STATE * HEADDIM);
        const float lam = lam_buf[bh * NCHUNK + c];
#pragma unroll
        for (int n = 0; n < D_STATE; ++n) {
            hprev_buf[base + n * HEADDIM + p] = hst[n];            // state entering chunk c
            hst[n] = lam * hst[n] + hc_buf[base + n * HEADDIM + p];
        }
    }
}

// ---- Phase C: Y += exp(cum_t) * C_t @ hprevT (64x32, K=32, WMMA) ----
__global__ __launch_bounds__(128)
void chunk_inter_kernel(const float* __restrict__ xbc,
                        const float* __restrict__ cume_buf,
                        const float* __restrict__ hprev_buf,
                        float* __restrict__ ybuf) {
    __shared__ float Ce[QCHUNK][33];   // exp(cum_t)*C[t][n]
    __shared__ float Hp[D_STATE][33];  // hprev[n][p]

    const int cid = blockIdx.x;
    const int c   = cid & (NCHUNK - 1);
    const int h   = (cid >> 5) & (NHEADS - 1);
    const int b   = cid >> 9;
    const int tokBase = b * LL + c * QCHUNK;

    const int tid  = threadIdx.x;
    const int lane = tid & 31;
    const int wave = tid >> 5;
    const int lh   = lane & 15;
    const int hi   = lane >> 4;

    for (int s = wave; s < QCHUNK; s += 4)
        Ce[s][lane] = xbc[(size_t)(tokBase + s) * CONV_DIM + D_INNER + D_STATE + lane] *
                      cume_buf[(size_t)cid * QCHUNK + s];
    for (int n = wave; n < D_STATE; n += 4)
        Hp[n][lane] = hprev_buf[(size_t)cid * (D_STATE * HEADDIM) + n * HEADDIM + lane];
    __syncthreads();

    v8f acc[2] = {v8f{}, v8f{}};
#pragma unroll
    for (int kk = 0; kk < D_STATE; kk += 4) {
        const int k0 = kk + hi * 2, k1 = k0 + 1;
        v2f a; a[0] = Ce[(wave << 4) + lh][k0]; a[1] = Ce[(wave << 4) + lh][k1];
#pragma unroll
        for (int j = 0; j < 2; ++j) {
            v2f bb; bb[0] = Hp[k0][(j << 4) + lh]; bb[1] = Hp[k1][(j << 4) + lh];
            acc[j] = WMMA4(a, bb, acc[j]);
        }
    }
#pragma unroll
    for (int j = 0; j < 2; ++j) {
        const int p = (j << 4) + lh;
#pragma unroll
        for (int r = 0; r < 8; ++r) {
            const int t = (wave << 4) + (hi << 3) + r;
            const size_t o = (size_t)(tokBase + t) * D_INNER + h * HEADDIM + p;
            ybuf[o] += acc[j][r];
        }
    }
}

// ---------------- y = rmsnorm(y * silu(z), nw), in place over D_INNER=512 ----------------
__global__ __launch_bounds__(32)
void gate_rms_kernel(float* __restrict__ ybuf, const float* __restrict__ zx,
                     const float* __restrict__ nw) {
    const int tok  = blockIdx.x;
    const int lane = threadIdx.x;
    float* yrow = ybuf + (size_t)tok * D_INNER;
    const float* zrow = zx + (size_t)tok * D_IN_PROJ;
    float v[16]; float ss = 0.0f;
#pragma unroll
    for (int i = 0; i < 16; ++i) {
        const int idx = lane + i * 32;
        const float g = yrow[idx] * siluf(zrow[idx]);
        v[i] = g; ss += g * g;
    }
    ss = waveRedSum(ss);
    const float inv = rsqrtf(ss * (1.0f / D_INNER) + EPSF);
#pragma unroll
    for (int i = 0; i < 16; ++i) {
        const int idx = lane + i * 32;
        yrow[idx] = v[i] * inv * nw[idx];
    }
}

// ---------------- h = rmsnorm(tmp + h, w), in place over HIDDEN=256 ----------------
__global__ __launch_bounds__(32)
void resadd_rms_kernel(float* __restrict__ hbuf, const float* __restrict__ tmp,
                       const float* __restrict__ w) {
    const int tok  = blockIdx.x;
    const int lane = threadIdx.x;
    float* hrow = hbuf + (size_t)tok * HIDDEN;
    const float* trow = tmp + (size_t)tok * HIDDEN;
    float v[8]; float ss = 0.0f;
#pragma unroll
    for (int i = 0; i < 8; ++i) {
        const int idx = lane + i * 32;
        const float s = hrow[idx] + trow[idx];
        v[i] = s; ss += s * s;
    }
    ss = waveRedSum(ss);
    const float inv = rsqrtf(ss * (1.0f / HIDDEN) + EPSF);
#pragma unroll
    for (int i = 0; i < 8; ++i) {
        const int idx = lane + i * 32;
        hrow[idx] = v[i] * inv * w[idx];
    }
}

// ---------------- layernorm over 256, src -> dst ----------------
__global__ __launch_bounds__(32)
void layernorm_kernel(const float* __restrict__ src, float* __restrict__ dst,
                      const float* __restrict__ w, const float* __restrict__ bia) {
    const int tok  = blockIdx.x;
    const int lane = threadIdx.x;
    const float* srow = src + (size_t)tok * HIDDEN;
    float v[8]; float s = 0.0f;
#pragma unroll
    for (int i = 0; i < 8; ++i) { v[i] = srow[lane + i * 32]; s += v[i]; }
    s = waveRedSum(s);
    const float mean = s * (1.0f / HIDDEN);
    float vs = 0.0f;
#pragma unroll
    for (int i = 0; i < 8; ++i) { const float d = v[i] - mean; vs += d * d; }
    vs = waveRedSum(vs);
    const float inv = rsqrtf(vs * (1.0f / HIDDEN) + EPSF);
    float* drow = dst + (size_t)tok * HIDDEN;
#pragma unroll
    for (int i = 0; i < 8; ++i) {
        const int idx = lane + i * 32;
        drow[idx] = (v[i] - mean) * inv * w[idx] + bia[idx];
    }
}

// ---------------- host side ----------------
extern "C" void kernel_launch(void* const* d_in, const int* in_sizes, int n_in,
                              void* d_out, int out_size, void* d_ws, size_t ws_size,
                              hipStream_t stream) {
    const float* x      = (const float*)d_in[0];
    const float* ip_w   = (const float*)d_in[1];
    const float* ip_b   = (const float*)d_in[2];
    const float* m_inw  = (const float*)d_in[3];
    const float* m_inb  = (const float*)d_in[4];
    const float* m_convw= (const float*)d_in[5];
    const float* m_convb= (const float*)d_in[6];
    const float* m_dtb  = (const float*)d_in[7];
    const float* m_Alog = (const float*)d_in[8];
    const float* m_D    = (const float*)d_in[9];
    const float* m_nw   = (const float*)d_in[10];
    const float* m_outw = (const float*)d_in[11];
    const float* m_outb = (const float*)d_in[12];
    const float* rms_w  = (const float*)d_in[13];
    const float* ln1_w  = (const float*)d_in[14];
    const float* ln1_b  = (const float*)d_in[15];
    const float* w1     = (const float*)d_in[16];
    const float* b1     = (const float*)d_in[17];
    const float* w2     = (const float*)d_in[18];
    const float* b2     = (const float*)d_in[19];
    const float* ln2_w  = (const float*)d_in[20];
    const float* ln2_b  = (const float*)d_in[21];
    float* out = (float*)d_out;

    // workspace carve-out (floats)
    float* ws = (float*)d_ws;
    float* hbuf    = ws;                                    // TOKENS*256
    float* zx      = hbuf    + (size_t)TOKENS * HIDDEN;     // TOKENS*1104
    float* xbc     = zx      + (size_t)TOKENS * D_IN_PROJ;  // TOKENS*576
    float* dtbuf   = xbc     + (size_t)TOKENS * CONV_DIM;   // TOKENS*16
    float* ybuf    = dtbuf   + (size_t)TOKENS * NHEADS;     // TOKENS*512
    float* tmp256  = ybuf    + (size_t)TOKENS * D_INNER;    // TOKENS*256
    float* hc_buf  = tmp256  + (size_t)TOKENS * HIDDEN;     // NCID*1024
    float* hprev   = hc_buf  + (size_t)NCID * 1024;         // NCID*1024
    float* cume    = hprev   + (size_t)NCID * 1024;         // NCID*64
    float* lam     = cume    + (size_t)NCID * QCHUNK;       // NCID

    const dim3 gBlk(128);
    const int mTiles = TOKENS / 64;  // 512

    // 1) h = x @ ip_w + ip_b
    gemm_wmma_f32<0><<<dim3((HIDDEN + 63) / 64, mTiles), gBlk, 0, stream>>>(
        x, ip_w, ip_b, hbuf, TOKENS, HIDDEN, INPUT_DIM);

    for (int l = 0; l < NLAYERS; ++l) {
        const float* inw   = m_inw   + (size_t)l * HIDDEN * D_IN_PROJ;
        const float* inb   = m_inb   + (size_t)l * D_IN_PROJ;
        const float* convw = m_convw + (size_t)l * CONV_DIM * KCONV;
        const float* convb = m_convb + (size_t)l * CONV_DIM;
        const float* dtb   = m_dtb   + (size_t)l * NHEADS;
        const float* Alog  = m_Alog  + (size_t)l * NHEADS;
        const float* Dp    = m_D     + (size_t)l * NHEADS;
        const float* nw    = m_nw    + (size_t)l * D_INNER;
        const float* outw  = m_outw  + (size_t)l * D_INNER * HIDDEN;
        const float* outb  = m_outb  + (size_t)l * HIDDEN;
        const float* rw    = rms_w   + (size_t)l * HIDDEN;

        // 2) zx = h @ inw + inb
        gemm_wmma_f32<0><<<dim3((D_IN_PROJ + 63) / 64, mTiles), gBlk, 0, stream>>>(
            hbuf, inw, inb, zx, TOKENS, D_IN_PROJ, HIDDEN);

        // 3) conv + silu ; dt softplus
        {
            const size_t n = (size_t)TOKENS * CONV_DIM;
            conv_silu_kernel<<<(unsigned)((n + 255) / 256), 256, 0, stream>>>(zx, convw, convb, xbc);
            dt_softplus_kernel<<<(TOKENS * NHEADS + 255) / 256, 256, 0, stream>>>(zx, dtb, dtbuf);
        }

        // 4) chunked selective scan (WMMA)
        chunk_intra_kernel<<<NCID, 128, 0, stream>>>(xbc, dtbuf, Alog, Dp,
                                                     ybuf, hc_buf, cume, lam);
        chunk_seq_kernel<<<BB * NHEADS, 32, 0, stream>>>(hc_buf, lam, hprev);
        chunk_inter_kernel<<<NCID, 128, 0, stream>>>(xbc, cume, hprev, ybuf);

        // 5) gated rmsnorm (in place on ybuf)
        gate_rms_kernel<<<TOKENS, 32, 0, stream>>>(ybuf, zx, nw);

        // 6) tmp256 = ybuf @ outw + outb
        gemm_wmma_f32<0><<<dim3((HIDDEN + 63) / 64, mTiles), gBlk, 0, stream>>>(
            ybuf, outw, outb, tmp256, TOKENS, HIDDEN, D_INNER);

        // 7) h = rmsnorm(tmp256 + h, rms_w[l])
        resadd_rms_kernel<<<TOKENS, 32, 0, stream>>>(hbuf, tmp256, rw);
    }

    // 8) ln1 -> tmp256
    layernorm_kernel<<<TOKENS, 32, 0, stream>>>(hbuf, tmp256, ln1_w, ln1_b);

    // 9) ybuf = gelu(tmp256 @ w1 + b1)
    gemm_wmma_f32<1><<<dim3((2 * HIDDEN + 63) / 64, mTiles), gBlk, 0, stream>>>(
        tmp256, w1, b1, ybuf, TOKENS, 2 * HIDDEN, HIDDEN);

    // 10) hbuf = ybuf @ w2 + b2
    gemm_wmma_f32<0><<<dim3((OUT_DIM + 63) / 64, mTiles), gBlk, 0, stream>>>(
        ybuf, w2, b2, hbuf, TOKENS, OUT_DIM, 2 * HIDDEN);

    // 11) out = layernorm(hbuf, ln2)
    layernorm_kernel<<<TOKENS, 32, 0, stream>>>(hbuf, out, ln2_w, ln2_b);

    (void)in_sizes; (void)n_in; (void)out_size; (void)ws_size;
}